// OPTAttention_49959059587478
// MI455X (gfx1250) — compile-verified
//
#include <hip/hip_runtime.h>
#include <math.h>

// ---------------- problem constants (match reference) ----------------
#define NUM_SEQS   32
#define EMBED      2048
#define NUM_HEADS  32
#define HEAD_DIM   64
#define PBLOCK     16          // paged-KV block size
#define MAX_BLOCKS 128
#define MAX_CTX    2048
#define RANK       16
#define QKV_N      (3 * EMBED) // 6144
#define ATT_SCALE  0.125f      // 64^-0.5
#define NEGF       (-1e30f)

typedef float v2f __attribute__((ext_vector_type(2)));
typedef float v8f __attribute__((ext_vector_type(8)));

#if defined(__has_builtin)
#if __has_builtin(__builtin_amdgcn_wmma_f32_16x16x4_f32)
#define HAVE_WMMA_F32X4 1
#endif
#endif

// ---------------------------------------------------------------------
// z[s][r] = sum_e x[s][e] * lora_a[map[s]][e][r]     (z is [32,16])
// one workgroup per sequence; 16 lanes per r, wave-shuffle reduce.
// ---------------------------------------------------------------------
__global__ void lora_z_kernel(const float* __restrict__ x,
                              const float* __restrict__ lora_a,
                              const int* __restrict__ amap,
                              float* __restrict__ z) {
  const int s = blockIdx.x;
  const int r = threadIdx.x >> 4;   // 0..15
  const int i = threadIdx.x & 15;   // 0..15
  const int ad = amap[s];
  const float* xs = x + (size_t)s * EMBED;
  const float* a  = lora_a + (size_t)ad * EMBED * RANK + r;
  float p = 0.f;
  for (int e = i; e < EMBED; e += 16)
    p += xs[e] * a[(size_t)e * RANK];
  // reduce over the 16 lanes of each half-wave (groups stay inside a wave32)
  for (int off = 8; off; off >>= 1)
    p += __shfl_xor(p, off, 32);
  if (i == 0) z[s * RANK + r] = p;
}

// ---------------------------------------------------------------------
// out[32][N] = A[32][2048] @ B[2048][N] + bias + (z @ lora_b[adapter])
// One wave computes one 16x16 tile with V_WMMA_F32_16X16X4_F32.
// Fragment layouts per CDNA5 ISA 7.12.2:
//   A 16x4 : VGPR0 = {K=0 | K=2}, VGPR1 = {K=1 | K=3} across lane halves
//   B 4x16 : VGPR0 = {K=0 | K=2}, VGPR1 = {K=1 | K=3}, N = lane&15
//   C/D    : VGPR j holds rows j (lanes 0-15) and j+8 (lanes 16-31)
// The weight matrix is streamed with unconditional *speculative* prefetch
// (ISA 10.5: speculative prefetch past the allocation is silently dropped),
// keeping the inner loop branch-free.
// ---------------------------------------------------------------------
template <int N>
__global__ void wmma_gemm_lora(const float* __restrict__ A,
                               const float* __restrict__ B,
                               const float* __restrict__ bias,
                               const float* __restrict__ lora_b,
                               const float* __restrict__ z,
                               const int* __restrict__ amap,
                               float* __restrict__ out) {
  const int lane  = threadIdx.x & 31;
  const int wid   = threadIdx.x >> 5;
  const int ntile = blockIdx.x * (blockDim.x >> 5) + wid;  // grid sized exactly
  const int mtile = blockIdx.y;                            // 0 or 1
  const int half  = lane >> 4;                             // 0|1
  const int mn    = lane & 15;
  const int arow  = mtile * 16 + mn;                       // row fed to A frag
  const int col   = ntile * 16 + mn;                       // column of B / out

  v8f acc = {0.f, 0.f, 0.f, 0.f, 0.f, 0.f, 0.f, 0.f};
  const float* ap = A + (size_t)arow * EMBED;

  for (int k = 0; k < EMBED; k += 4) {
#if defined(HAVE_WMMA_F32X4)
    v2f a, b;
    a.x = ap[k + 2 * half];
    a.y = ap[k + 2 * half + 1];
    const float* bp = B + (size_t)(k + 2 * half) * N + col;
    b.x = bp[0];
    b.y = bp[(size_t)N];
    // branch-free speculative prefetch 16 rows ahead of the weight stream
    __builtin_prefetch(B + (size_t)(k + 16 + 2 * half) * N + col, 0, 1);
    acc = __builtin_amdgcn_wmma_f32_16x16x4_f32(
        /*neg_a=*/false, a, /*neg_b=*/false, b,
        /*c_mod=*/(short)0, acc, /*reuse_a=*/false, /*reuse_b=*/false);
#else
    for (int kk = 0; kk < 4; ++kk) {
      const float bv = B[(size_t)(k + kk) * N + col];
      for (int j = 0; j < 8; ++j)
        acc[j] += A[(size_t)(mtile * 16 + j + 8 * half) * EMBED + k + kk] * bv;
    }
#endif
  }

  // epilogue: bias + rank-16 LoRA correction (lora_b stays resident in L2)
  const float bcol = bias[col];
  for (int j = 0; j < 8; ++j) {
    const int orow = mtile * 16 + j + 8 * half;
    const int ad   = amap[orow];
    const float* zr = z + orow * RANK;
    const float* lb = lora_b + (size_t)ad * RANK * N + col;
    float l = 0.f;
    for (int r = 0; r < RANK; ++r)
      l += zr[r] * lb[(size_t)r * N];
    out[(size_t)orow * N + col] = acc[j] + bcol + l;
  }
}

// ---------------------------------------------------------------------
// Flash-decode paged attention. One workgroup per (head, seq).
// 8 waves stride over tokens; 2 dims/lane; online softmax; LDS merge.
// Token ctx-1 reads the freshly projected K/V (cache inputs are immutable;
// block_tables is a permutation, so that slot belongs only to this seq).
// ---------------------------------------------------------------------
__global__ void paged_attn_kernel(const float* __restrict__ qkv,
                                  const float* __restrict__ kcache,
                                  const float* __restrict__ vcache,
                                  const int* __restrict__ btab,
                                  const int* __restrict__ clen,
                                  float* __restrict__ attn) {
  const int h = blockIdx.x;
  const int s = blockIdx.y;
  const int lane = threadIdx.x & 31;
  const int wid  = threadIdx.x >> 5;
  const int ctx  = clen[s];

  const float* qv = qkv + (size_t)s * QKV_N + h * HEAD_DIM;
  const float* nk = qv + EMBED;       // new key   for this token
  const float* nv = qv + 2 * EMBED;   // new value for this token
  const float2 q = *(const float2*)(qv + 2 * lane);
  const int* bt = btab + s * MAX_BLOCKS;

  float m = NEGF, l = 0.f;
  float2 acc = {0.f, 0.f};

  for (int t = wid; t < ctx; t += 8) {
    const float *kp, *vp;
    if (t == ctx - 1) {
      kp = nk; vp = nv;
    } else {
      const int blk = bt[t >> 4];
      const size_t base =
          (((size_t)blk * PBLOCK + (t & (PBLOCK - 1))) * NUM_HEADS + h) * HEAD_DIM;
      kp = kcache + base; vp = vcache + base;
    }
    const float2 kk = *(const float2*)(kp + 2 * lane);
    float p = q.x * kk.x + q.y * kk.y;
    for (int off = 16; off; off >>= 1)
      p += __shfl_xor(p, off, 32);
    const float sc = p * ATT_SCALE;

    const float nm = fmaxf(m, sc);
    const float co = __expf(m - nm);
    const float w  = __expf(sc - nm);
    const float2 vv = *(const float2*)(vp + 2 * lane);
    acc.x = acc.x * co + w * vv.x;
    acc.y = acc.y * co + w * vv.y;
    l = l * co + w;
    m = nm;
  }

  __shared__ float sm[8], sl[8], sacc[8][HEAD_DIM];
  sacc[wid][2 * lane]     = acc.x;
  sacc[wid][2 * lane + 1] = acc.y;
  if (lane == 0) { sm[wid] = m; sl[wid] = l; }
  __syncthreads();

  if (threadIdx.x < HEAD_DIM) {
    float M = NEGF;
    for (int w2 = 0; w2 < 8; ++w2) M = fmaxf(M, sm[w2]);
    float L = 0.f, o = 0.f;
    for (int w2 = 0; w2 < 8; ++w2) {
      const float c = __expf(sm[w2] - M);
      L += c * sl[w2];
      o += c * sacc[w2][threadIdx.x];
    }
    attn[(size_t)s * EMBED + h * HEAD_DIM + threadIdx.x] = o / L;
  }
}

// ---------------------------------------------------------------------
extern "C" void kernel_launch(void* const* d_in, const int* in_sizes, int n_in,
                              void* d_out, int out_size, void* d_ws, size_t ws_size,
                              hipStream_t stream) {
  (void)in_sizes; (void)n_in; (void)out_size; (void)ws_size;
  const float* hs     = (const float*)d_in[0];
  const float* w_qkv  = (const float*)d_in[1];
  const float* b_qkv  = (const float*)d_in[2];
  const float* w_out  = (const float*)d_in[3];
  const float* b_out  = (const float*)d_in[4];
  const float* qla    = (const float*)d_in[5];
  const float* qlb    = (const float*)d_in[6];
  const float* ola    = (const float*)d_in[7];
  const float* olb    = (const float*)d_in[8];
  const float* kcache = (const float*)d_in[9];
  const float* vcache = (const float*)d_in[10];
  const int*   btab   = (const int*)d_in[11];
  const int*   clen   = (const int*)d_in[12];
  const int*   amap   = (const int*)d_in[13];
  float* out = (float*)d_out;

  float* ws   = (float*)d_ws;
  float* qkv  = ws;                           // 32*6144
  float* attn = qkv  + NUM_SEQS * QKV_N;      // 32*2048
  float* z1   = attn + NUM_SEQS * EMBED;      // 32*16
  float* z2   = z1   + NUM_SEQS * RANK;       // 32*16

  // 1) LoRA-A contraction for QKV
  lora_z_kernel<<<NUM_SEQS, 256, 0, stream>>>(hs, qla, amap, z1);

  // 2) fused QKV projection (WMMA) + bias + LoRA-B
  {
    dim3 grid(QKV_N / 16 / 4, 2);             // 96 x 2, 4 waves/block
    wmma_gemm_lora<QKV_N><<<grid, 128, 0, stream>>>(
        hs, w_qkv, b_qkv, qlb, z1, amap, qkv);
  }

  // 3) paged flash-decode attention (HBM-bandwidth dominated)
  paged_attn_kernel<<<dim3(NUM_HEADS, NUM_SEQS), 256, 0, stream>>>(
      qkv, kcache, vcache, btab, clen, attn);

  // 4) LoRA-A contraction for output projection
  lora_z_kernel<<<NUM_SEQS, 256, 0, stream>>>(attn, ola, amap, z2);

  // 5) fused output projection (WMMA) + bias + LoRA-B
  {
    dim3 grid(EMBED / 16 / 4, 2);             // 32 x 2
    wmma_gemm_lora<EMBED><<<grid, 128, 0, stream>>>(
        attn, w_out, b_out, olb, z2, amap, out);
  }
}